// NonAutoRegression_47175920779374
// MI455X (gfx1250) — compile-verified
//
#include <hip/hip_runtime.h>
#include <math.h>

#define BB   8
#define OBJ  32
#define HIST 512
#define DM   128
#define PRED 512

typedef __attribute__((ext_vector_type(2))) float v2f;
typedef __attribute__((ext_vector_type(8))) float v8f;

// ---------------------------------------------------------------------------
// Kernel 1: fw[f] = pe[f,:] . w_futr + b_futr   (sinusoidal PE on the fly)
// ---------------------------------------------------------------------------
__global__ void fw_kernel(const float* __restrict__ w_futr,
                          const float* __restrict__ b_futr,
                          float* __restrict__ fw) {
    int f = blockIdx.x * blockDim.x + threadIdx.x;
    if (f >= PRED) return;
    const float cc = -logf(10000.0f) / (float)DM;
    float acc = 0.0f;
    #pragma unroll 8
    for (int i = 0; i < DM / 2; ++i) {
        float div = expf((float)(2 * i) * cc);
        float ang = (float)f * div;
        acc += sinf(ang) * w_futr[2 * i] + cosf(ang) * w_futr[2 * i + 1];
    }
    fw[f] = acc + b_futr[0];
}

// ---------------------------------------------------------------------------
// Kernel 2: per (b,o): ctx[d] = sum_h ew[h] * relu(hist[h,:] @ W^T + b)[d]
// One block per (b,o); 8 waves; wave w owns output columns e0 = 16*w .. +15.
// fp32 WMMA 16x16x4, A staged in LDS, B (W_tgt slice) held in registers.
// ---------------------------------------------------------------------------
__global__ void __launch_bounds__(256)
ctx_kernel(const float* __restrict__ hist,
           const float* __restrict__ mem,
           const float* __restrict__ Wt,      // [DM, DM], row-major (e, d)
           const float* __restrict__ b_tgt,   // [DM]
           const float* __restrict__ w_enc,   // [DM]
           const float* __restrict__ b_enc,   // [1]
           float* __restrict__ ctx) {         // [BB*OBJ, DM]
    __shared__ float ew[HIST];
    __shared__ float Atile[16 * 132];          // 16 rows, pitch 132 (pad vs 64 banks)

    const int    bo   = blockIdx.x;
    const int    t    = threadIdx.x;
    const size_t base = (size_t)bo * HIST * DM;

    // ---- ew for this block's 512 rows ----
    {
        const float be = b_enc[0];
        const float4* wv = (const float4*)w_enc;
        for (int r = t; r < HIST; r += 256) {
            const float4* mr = (const float4*)(mem + base + (size_t)r * DM);
            float acc = 0.0f;
            #pragma unroll
            for (int j = 0; j < DM / 4; ++j) {
                float4 a = mr[j], b = wv[j];
                acc += a.x * b.x + a.y * b.y + a.z * b.z + a.w * b.w;
            }
            ew[r] = acc + be;
        }
    }

    const int lane = t & 31;
    const int wave = t >> 5;
    const int n    = lane & 15;          // A-row m and B-col n (both = lane%16)
    const int kb   = (lane >> 4) << 1;   // K sub-offset: 0 for lanes 0-15, 2 for 16-31
    const int e0   = wave * 16;
    const int mb   = (lane >> 4) * 8;    // C/D: vgpr i holds M = mb + i

    // ---- B fragments: W_tgt[e0+n, :] packed as 32 x v2f (K=4 steps) ----
    v2f Bf[32];
    #pragma unroll
    for (int kk = 0; kk < 32; ++kk) {
        const float* p = Wt + (size_t)(e0 + n) * DM + kk * 4 + kb;
        Bf[kk].x = p[0];
        Bf[kk].y = p[1];
    }
    const float bias = b_tgt[e0 + n];
    float ctxacc = 0.0f;

    __syncthreads();   // ew ready

    for (int ht = 0; ht < HIST / 16; ++ht) {
        const int h0 = ht * 16;
        // ---- stage A tile [16 x 128] into LDS (512 float4, 2 per thread) ----
        {
            const float4* src = (const float4*)(hist + base + (size_t)h0 * DM);
            #pragma unroll
            for (int q = 0; q < 2; ++q) {
                int idx = t + q * 256;           // 0..511
                int row = idx >> 5;              // /32
                int c4  = idx & 31;
                float4 v = src[row * 32 + c4];
                *(float4*)(&Atile[row * 132 + c4 * 4]) = v;  // pitch 528B, 16B aligned
            }
        }
        __syncthreads();

        // ---- 16x16 output tile: 32 K-steps of fp32 WMMA ----
        v8f c = {0.f, 0.f, 0.f, 0.f, 0.f, 0.f, 0.f, 0.f};
        #pragma unroll
        for (int kk = 0; kk < 32; ++kk) {
            v2f a;
            const float* ap = &Atile[n * 132 + kk * 4 + kb];
            a.x = ap[0];
            a.y = ap[1];
            c = __builtin_amdgcn_wmma_f32_16x16x4_f32(
                    /*neg_a=*/false, a, /*neg_b=*/false, Bf[kk],
                    /*c_mod=*/(short)0, c, /*reuse_a=*/false, /*reuse_b=*/false);
        }

        // ---- bias + relu + ew-weighted reduction over the 8 M rows here ----
        #pragma unroll
        for (int i = 0; i < 8; ++i) {
            float y = c[i] + bias;
            y = y > 0.0f ? y : 0.0f;
            ctxacc += y * ew[h0 + mb + i];
        }
        __syncthreads();   // before next tile overwrites Atile
    }

    // combine the two half-wave partials for the same e = e0 + n
    ctxacc += __shfl_xor(ctxacc, 16, 32);
    if (lane < 16) ctx[(size_t)bo * DM + e0 + n] = ctxacc;
}

// ---------------------------------------------------------------------------
// Kernel 3: out[bo, f, d] = fw[f] * ctx[bo, d]   (float4 over 64 MB)
// ---------------------------------------------------------------------------
__global__ void expand_kernel(const float* __restrict__ fw,
                              const float* __restrict__ ctx,
                              float* __restrict__ out) {
    const size_t total4 = (size_t)BB * OBJ * PRED * DM / 4;
    size_t v = (size_t)blockIdx.x * blockDim.x + threadIdx.x;
    if (v >= total4) return;
    int d4 = (int)(v & 31);                 // DM/4 = 32
    int f  = (int)((v >> 5) & (PRED - 1));
    int bo = (int)(v >> 14);                // / (32*512)
    float  s  = fw[f];
    float4 cv = ((const float4*)ctx)[bo * 32 + d4];
    float4 o;
    o.x = s * cv.x; o.y = s * cv.y; o.z = s * cv.z; o.w = s * cv.w;
    ((float4*)out)[v] = o;
}

// ---------------------------------------------------------------------------
extern "C" void kernel_launch(void* const* d_in, const int* in_sizes, int n_in,
                              void* d_out, int out_size, void* d_ws, size_t ws_size,
                              hipStream_t stream) {
    const float* hist = (const float*)d_in[0];  // [B,OBJ,HIST,D]
    const float* mem  = (const float*)d_in[1];  // [B,OBJ,HIST,D]
    const float* Wt   = (const float*)d_in[2];  // [D,D]
    const float* btg  = (const float*)d_in[3];  // [D]
    const float* wen  = (const float*)d_in[4];  // [D]
    const float* ben  = (const float*)d_in[5];  // [1]
    const float* wfu  = (const float*)d_in[6];  // [D]
    const float* bfu  = (const float*)d_in[7];  // [1]
    float* out = (float*)d_out;

    float* fw  = (float*)d_ws;        // 512 floats
    float* ctx = fw + PRED;           // 256*128 floats

    fw_kernel<<<(PRED + 255) / 256, 256, 0, stream>>>(wfu, bfu, fw);
    ctx_kernel<<<BB * OBJ, 256, 0, stream>>>(hist, mem, Wt, btg, wen, ben, ctx);

    const size_t total4 = (size_t)BB * OBJ * PRED * DM / 4;
    expand_kernel<<<(unsigned)((total4 + 255) / 256), 256, 0, stream>>>(fw, ctx, out);
}